// SSIMLoss_32916629357047
// MI455X (gfx1250) — compile-verified
//
#include <hip/hip_runtime.h>

// ---------------------------------------------------------------------------
// SSIM loss for 32x1x512x512 fp32 via CDNA5 V_WMMA_F32_16X16X4_F32.
//
// Separable 11x11 Gaussian conv as WMMA matmuls per 16x16 output tile:
//   H(26x16)  = In(26x28) x Wh(28x16),  Wh[K][c] = g[K-c] (band, zero-padded K)
//   Out(16x16)= Aw(16x28) x H(28x16),   Aw[m][K] = g[K-m] = Wh[K][m]
// One wave (32 lanes) per tile; 5 conv quantities (x, y, x^2, y^2, xy);
// 7 K-chunks of 4 per pass => 5*(7+7+7) = 105 WMMAs per tile.
//
// Layout assumptions (from CDNA5 ISA 7.12.2):
//   A 16x4 f32 : lane L -> M = L%16; VGPR0 holds K = 4k+(L<16?0:2), VGPR1 K+1
//   B 4x16 f32 : lane L -> N = L%16; VGPR0 holds K = 4k+(L<16?0:2), VGPR1 K+1
//   C/D 16x16  : vgpr v, lane L -> (M = v + 8*(L/16), N = L%16)
// ---------------------------------------------------------------------------

typedef __attribute__((ext_vector_type(2))) float v2f;
typedef __attribute__((ext_vector_type(8))) float v8f;

#define WAVES_PER_BLOCK 8
#define RW 26   // halo region rows  (16 + 2*5)
#define CW 28   // halo region cols, K padded to multiple of 4

__device__ __forceinline__ v8f wmma4(v2f a, v2f b, v8f c) {
  // D = A(16x4) * B(4x16) + C, fp32 everywhere (matches reference precision)
  return __builtin_amdgcn_wmma_f32_16x16x4_f32(
      /*neg_a=*/false, a, /*neg_b=*/false, b,
      /*c_mod=*/(short)0, c, /*reuse_a=*/false, /*reuse_b=*/false);
}

__device__ __forceinline__ v8f zero8() {
  v8f z = {0.f, 0.f, 0.f, 0.f, 0.f, 0.f, 0.f, 0.f};
  return z;
}

// Unnormalized Gaussian (faithful to torch module): g[d] = exp(-d^2/4)
__device__ __forceinline__ float gweight(int d) {
  // exp(-d^2/4) = exp2(-d^2 * log2(e)/4)
  return ((unsigned)d <= 10u)
             ? exp2f(-(float)(d * d) * 0.36067376022224085f)
             : 0.0f;
}

// Conv input per quantity (q is a compile-time constant after unrolling)
__device__ __forceinline__ float qv(int q, float x, float y) {
  switch (q) {
    case 0:  return x;
    case 1:  return y;
    case 2:  return x * x;
    case 3:  return y * y;
    default: return x * y;
  }
}

__global__ __launch_bounds__(256) void ssim_tile_kernel(
    const float* __restrict__ img1, const float* __restrict__ img2,
    float* __restrict__ partial) {
  __shared__ float s_x[WAVES_PER_BLOCK][RW * CW];
  __shared__ float s_y[WAVES_PER_BLOCK][RW * CW];
  __shared__ float s_h[WAVES_PER_BLOCK][28 * 16];  // H staging (rows 26,27 = 0)
  __shared__ float s_red[WAVES_PER_BLOCK];

  const int lane = threadIdx.x & 31;
  const int wv   = threadIdx.x >> 5;

  // 32 images * 32x32 tiles, one wave per tile
  const int tileId = blockIdx.x * WAVES_PER_BLOCK + wv;
  const int n  = tileId >> 10;
  const int tr = ((tileId >> 5) & 31) << 4;
  const int tc = (tileId & 31) << 4;

  const float* p1 = img1 + (size_t)n * 512 * 512;
  const float* p2 = img2 + (size_t)n * 512 * 512;

  float* wx = s_x[wv];
  float* wy = s_y[wv];
  float* wh = s_h[wv];

  const int col = lane & 15;
  const int kof = (lane & 16) ? 2 : 0;

  // Zero the K-padding rows (26,27) of the H buffer once.
  wh[(26 + (lane >> 4)) * 16 + col] = 0.0f;

  // Stage 26x28 halo regions of both images (zero-padded like lax.conv).
  for (int i = lane; i < RW * CW; i += 32) {
    const int r = i / CW, c = i % CW;
    const int gr = tr - 5 + r, gc = tc - 5 + c;
    float a = 0.f, b = 0.f;
    if (gr >= 0 && gr < 512 && gc >= 0 && gc < 512) {
      const size_t off = (size_t)gr * 512 + gc;
      a = p1[off];
      b = p2[off];
    }
    wx[i] = a;
    wy[i] = b;
  }
  // Same-wave DS ops are in-order (ISA 7.3): no barrier needed — all LDS
  // buffers here are wave-private.

  // Per-lane Gaussian band-matrix operand, shared by both passes:
  // value(lane, k, j) = g[(4k + kof + j) - (lane&15)]
  v2f w7[7];
#pragma unroll
  for (int k = 0; k < 7; ++k) {
    const int j0 = 4 * k + kof;
    w7[k].x = gweight(j0 - col);
    w7[k].y = gweight(j0 + 1 - col);
  }

  v8f acc[5];
#pragma unroll
  for (int q = 0; q < 5; ++q) {
    // ---- Horizontal pass: rows 0..15 (top) and 10..25 (bottom) ----
    v8f htop = zero8(), hbot = zero8();
    const int rt = lane & 15;
    const int rb = rt + 10;
#pragma unroll
    for (int k = 0; k < 7; ++k) {
      const int K0 = 4 * k + kof;
      const float xt0 = wx[rt * CW + K0], xt1 = wx[rt * CW + K0 + 1];
      const float yt0 = wy[rt * CW + K0], yt1 = wy[rt * CW + K0 + 1];
      const float xb0 = wx[rb * CW + K0], xb1 = wx[rb * CW + K0 + 1];
      const float yb0 = wy[rb * CW + K0], yb1 = wy[rb * CW + K0 + 1];
      v2f at, ab;
      at.x = qv(q, xt0, yt0);
      at.y = qv(q, xt1, yt1);
      ab.x = qv(q, xb0, yb0);
      ab.y = qv(q, xb1, yb1);
      htop = wmma4(at, w7[k], htop);
      hbot = wmma4(ab, w7[k], hbot);
    }

    // ---- Scatter H (C/D layout) into row-major LDS [28][16] ----
    const int mBase = (lane & 16) ? 8 : 0;
#pragma unroll
    for (int v = 0; v < 8; ++v) {
      wh[(mBase + v) * 16 + col]      = htop[v];
      wh[(mBase + v + 10) * 16 + col] = hbot[v];  // rows 10..25
    }

    // ---- Vertical pass: Out = Aw x H ----
    v8f o = zero8();
#pragma unroll
    for (int k = 0; k < 7; ++k) {
      const int r0 = 4 * k + kof;
      v2f b;
      b.x = wh[r0 * 16 + col];
      b.y = wh[(r0 + 1) * 16 + col];
      o = wmma4(w7[k], b, o);
    }
    acc[q] = o;
  }

  // ---- Elementwise SSIM map + local accumulation ----
  const float C1 = 6.5025f, C2 = 58.5225f;
  float lsum = 0.f;
#pragma unroll
  for (int v = 0; v < 8; ++v) {
    const float mu1 = acc[0][v], mu2 = acc[1][v];
    const float cxx = acc[2][v], cyy = acc[3][v], cxy = acc[4][v];
    const float mu1s = mu1 * mu1, mu2s = mu2 * mu2, mu12 = mu1 * mu2;
    const float s1 = cxx - mu1s, s2 = cyy - mu2s, s12 = cxy - mu12;
    const float num = (2.f * mu12 + C1) * (2.f * s12 + C2);
    const float den = (mu1s + mu2s + C1) * (s1 + s2 + C2);
    lsum += (1.f - num / den) * 0.5f;
  }

  // ---- Wave (32) -> workgroup -> per-block partial ----
#pragma unroll
  for (int o = 16; o > 0; o >>= 1) lsum += __shfl_xor(lsum, o, 32);
  if (lane == 0) s_red[wv] = lsum;
  __syncthreads();
  if (threadIdx.x == 0) {
    float t = 0.f;
#pragma unroll
    for (int i = 0; i < WAVES_PER_BLOCK; ++i) t += s_red[i];
    partial[blockIdx.x] = t;
  }
}

// Deterministic single-block final reduction: mean over 32*512*512 pixels.
__global__ __launch_bounds__(256) void ssim_reduce_kernel(
    const float* __restrict__ partial, int n, float* __restrict__ out) {
  __shared__ float s[256];
  float t = 0.f;
  for (int i = threadIdx.x; i < n; i += 256) t += partial[i];
  s[threadIdx.x] = t;
  __syncthreads();
  for (int off = 128; off > 0; off >>= 1) {
    if (threadIdx.x < off) s[threadIdx.x] += s[threadIdx.x + off];
    __syncthreads();
  }
  if (threadIdx.x == 0)
    out[0] = s[0] * (1.0f / (32.0f * 512.0f * 512.0f));
}

extern "C" void kernel_launch(void* const* d_in, const int* in_sizes, int n_in,
                              void* d_out, int out_size, void* d_ws,
                              size_t ws_size, hipStream_t stream) {
  const float* img1 = (const float*)d_in[0];
  const float* img2 = (const float*)d_in[1];
  float* partial = (float*)d_ws;  // 4096 floats = 16 KB of scratch

  // 32 images * (512/16)^2 tiles = 32768 waves; 8 waves per 256-thread block.
  const int nblocks = 32 * 32 * 32 / WAVES_PER_BLOCK;  // 4096
  ssim_tile_kernel<<<nblocks, 256, 0, stream>>>(img1, img2, partial);
  ssim_reduce_kernel<<<1, 256, 0, stream>>>(partial, nblocks, (float*)d_out);
}